// DSASparseAttentionCP_40372692583237
// MI455X (gfx1250) — compile-verified
//
#include <hip/hip_runtime.h>
#include <hip/hip_bf16.h>
#include <math.h>

// Problem constants (match the reference).
constexpr int B_    = 1;
constexpr int H_    = 16;
constexpr int SQ_   = 1024;
constexpr int SKV_  = 4096;
constexpr int D_    = 128;
constexpr int TOPK_ = 64;
constexpr float SCALE_ = 0.08838834764831845f; // 128^-0.5

typedef __attribute__((ext_vector_type(2))) float v2f;
typedef __attribute__((ext_vector_type(4))) float v4f;
typedef __attribute__((ext_vector_type(8))) float v8f;

// One block of 128 threads (4 wave32) per (h, s).
//  - QK: each wave computes scores for one 16-row topk tile with
//    v_wmma_f32_16x16x4_f32, contracting over the 128-dim in 32 steps.
//    A = gathered K rows (16x4 per step), B = q chunk broadcast over N.
//  - softmax * topk_scores with deferred normalization.
//  - PV: each wave owns 16 topk rows; one global_load_b128 per lane covers a
//    full 512B V row; 2KB LDS cross-wave reduction produces the output.
__global__ __launch_bounds__(128)
void dsa_sparse_attn_kernel(const float* __restrict__ q,
                            const float* __restrict__ k,
                            const float* __restrict__ v,
                            const int*   __restrict__ topk_idx,
                            const float* __restrict__ topk_sc,
                            float*       __restrict__ out)
{
    const int s    = blockIdx.x;     // query position
    const int h    = blockIdx.y;     // head
    const int tid  = threadIdx.x;    // 0..127
    const int lane = tid & 31;
    const int wave = tid >> 5;       // 0..3 -> topk tile

    __shared__ int   s_idx[TOPK_];
    __shared__ float s_ts [TOPK_];
    __shared__ float s_q  [D_];
    __shared__ float s_sc [TOPK_];
    __shared__ float s_w  [TOPK_];
    __shared__ float s_red[TOPK_];
    __shared__ float s_pv [4][D_];   // per-wave PV partials
    __shared__ float s_sum;

    if (tid < TOPK_) {
        s_idx[tid] = topk_idx[s * TOPK_ + tid];
        s_ts [tid] = topk_sc [s * TOPK_ + tid];
    }
    // 128 threads load the 128-dim query row.
    s_q[tid] = q[((size_t)h * SQ_ + s) * D_ + tid];
    __syncthreads();

    // ---------------- QK scores via WMMA f32 16x16x4 ----------------
    {
        // A-matrix 16x4 f32 layout: lanes 0-15 hold (M=lane, K=0/1),
        // lanes 16-31 hold (M=lane-16, K=2/3).
        const int m    = wave * 16 + (lane & 15);         // topk row this lane feeds
        const int doff = (lane >> 4) * 2;                 // 0 or 2 within 4-dim chunk
        const float* __restrict__ krow =
            k + ((size_t)h * SKV_ + s_idx[m]) * (size_t)D_;

        v8f c = {};
        #pragma unroll
        for (int step = 0; step < D_ / 4; ++step) {
            const int d0 = step * 4 + doff;               // 8B-aligned
            v2f a = *(const v2f*)(krow + d0);             // gathered K chunk
            v2f b = { s_q[d0], s_q[d0 + 1] };             // q chunk, same for all N
            c = __builtin_amdgcn_wmma_f32_16x16x4_f32(
                    false, a, false, b, (short)0, c, false, false);
        }
        // C layout: VGPR r, lanes 0-15 -> (M=r, N=lane); lanes 16-31 -> (M=8+r).
        // All N columns are identical (B columns identical), so lane 0 / lane 16
        // hold the full tile of scores.
        if (lane == 0 || lane == 16) {
            const int mbase = wave * 16 + (lane >> 4) * 8;
            #pragma unroll
            for (int r = 0; r < 8; ++r)
                s_sc[mbase + r] = c[r] * SCALE_;
        }
    }
    __syncthreads();

    // ---------------- softmax * topk_scores (deferred norm) ----------------
    if (tid < TOPK_) s_red[tid] = s_sc[tid];
    __syncthreads();
    #pragma unroll
    for (int off = 32; off > 0; off >>= 1) {
        if (tid < off) s_red[tid] = fmaxf(s_red[tid], s_red[tid + off]);
        __syncthreads();
    }
    const float mx = s_red[0];
    __syncthreads();
    if (tid < TOPK_) {
        const float e = __expf(s_sc[tid] - mx) * s_ts[tid];
        s_w  [tid] = e;
        s_red[tid] = e;
    }
    __syncthreads();
    #pragma unroll
    for (int off = 32; off > 0; off >>= 1) {
        if (tid < off) s_red[tid] += s_red[tid + off];
        __syncthreads();
    }
    if (tid == 0) s_sum = s_red[0];
    __syncthreads();

    // ---------------- PV: one b128 per lane covers a full 512B V row -------
    {
        v4f acc = {0.0f, 0.0f, 0.0f, 0.0f};
        const int dbase = lane * 4;                       // 4 dims per lane
        #pragma unroll
        for (int tt = 0; tt < TOPK_ / 4; ++tt) {
            const int t = wave * (TOPK_ / 4) + tt;        // this wave's topk rows
            const float* __restrict__ vrow =
                v + ((size_t)h * SKV_ + s_idx[t]) * (size_t)D_;
            v4f val = *(const v4f*)(vrow + dbase);        // 32 lanes x 16B = 512B
            acc += val * s_w[t];
        }
        *(v4f*)(&s_pv[wave][dbase]) = acc;
    }
    __syncthreads();

    // Cross-wave reduction + deferred softmax normalization, 1 dim per thread.
    {
        const float rs = 1.0f / (s_sum + 1e-12f);
        const float total = s_pv[0][tid] + s_pv[1][tid] + s_pv[2][tid] + s_pv[3][tid];
        out[((size_t)h * SQ_ + s) * D_ + tid] = total * rs;
    }
}

extern "C" void kernel_launch(void* const* d_in, const int* in_sizes, int n_in,
                              void* d_out, int out_size, void* d_ws, size_t ws_size,
                              hipStream_t stream) {
    (void)in_sizes; (void)n_in; (void)out_size; (void)d_ws; (void)ws_size;
    const float* q  = (const float*)d_in[0];
    const float* k  = (const float*)d_in[1];
    const float* v  = (const float*)d_in[2];
    const int*   ti = (const int*)  d_in[3];
    const float* ts = (const float*)d_in[4];
    float* out = (float*)d_out;

    dim3 grid(SQ_, H_, 1);
    dim3 block(128, 1, 1);
    hipLaunchKernelGGL(dsa_sparse_attn_kernel, grid, block, 0, stream,
                       q, k, v, ti, ts, out);
}